// BotRGCN_70729521430927
// MI455X (gfx1250) — compile-verified
//
#include <hip/hip_runtime.h>

typedef __attribute__((ext_vector_type(2))) float v2f;
typedef __attribute__((ext_vector_type(4))) float v4f;
typedef __attribute__((ext_vector_type(8))) float v8f;
typedef __attribute__((ext_vector_type(4))) int v4i;

namespace {
constexpr int kN = 100000;     // nodes
constexpr int kE = 3200000;    // edges
constexpr int kH = 64;         // hidden
constexpr float kNeg = 0.01f;  // leaky slope
constexpr float kEps = 1e-5f;  // BN eps
constexpr int kNT = kN / 16;   // 6250 row tiles (exact)
constexpr int kKC = 64;        // K-chunk (floats) staged per async step
constexpr int kNC = 768 / kKC; // 12 chunks

// workspace layout (float elements)
constexpr size_t WS_AC  = 0;                        // folded BN scale A[3][64], shift C[3][64]
constexpr size_t WS_XR  = 512;                      // xr  [2][N][64]
constexpr size_t WS_S   = WS_XR + (size_t)2*kN*kH;  // s   [2][N][64] (segment sums)
constexpr size_t WS_CNT = WS_S  + (size_t)2*kN*kH;  // cnt [2][N]
constexpr size_t WS_X2  = WS_CNT + (size_t)2*kN;    // x2  [N][64]
}

__device__ __forceinline__ float leaky(float v) { return v > 0.0f ? v : kNeg * v; }

// fp32 WMMA, D = A(16x4) * B(4x16) + C, exact fp32 path on the matrix pipe.
__device__ __forceinline__ v8f wmma_f32_k4(v2f a, v2f b, v8f c) {
  return __builtin_amdgcn_wmma_f32_16x16x4_f32(false, a, false, b, (short)0, c,
                                               false, false);
}

// --- CDNA5 async Global->LDS (ASYNCcnt-tracked, no VGPR round trip) --------
#if __has_builtin(__builtin_amdgcn_global_load_async_to_lds_b128)
#define ASYNC_LDS 1
#else
#define ASYNC_LDS 0
#endif

__device__ __forceinline__ void copy16_to_lds(const float* g, float* l) {
#if ASYNC_LDS
  __builtin_amdgcn_global_load_async_to_lds_b128(
      (v4i __attribute__((address_space(1)))*)g,
      (v4i __attribute__((address_space(3)))*)l, 0, 0);
#else
  *(v4f*)l = *(const v4f*)g;  // synchronous fallback (still coalesced)
#endif
}

template <int N>
__device__ __forceinline__ void wait_async_lds() {
#if ASYNC_LDS
#if __has_builtin(__builtin_amdgcn_s_wait_asynccnt)
  __builtin_amdgcn_s_wait_asynccnt(N);
#else
  asm volatile("s_wait_asynccnt %0" ::"n"(N) : "memory");
#endif
#endif
}

// ---------------------------------------------------------------------------
// Fold Linear bias + BatchNorm (eval) into per-channel scale/shift:
//   y = dot * A + C,  A = g*rsqrt(v+eps),  C = b*A + beta - m*A
// ---------------------------------------------------------------------------
__global__ void bn_fold_kernel(
    const float* __restrict__ desB, const float* __restrict__ desG,
    const float* __restrict__ desBe, const float* __restrict__ desM,
    const float* __restrict__ desV,
    const float* __restrict__ numB, const float* __restrict__ numG,
    const float* __restrict__ numBe, const float* __restrict__ numM,
    const float* __restrict__ numV,
    const float* __restrict__ catB, const float* __restrict__ catG,
    const float* __restrict__ catBe, const float* __restrict__ catM,
    const float* __restrict__ catV,
    float* __restrict__ ac) {
  int t = threadIdx.x;
  if (t >= 3 * kH) return;
  int br = t / kH, n = t % kH;
  const float* b  = br == 0 ? desB  : (br == 1 ? numB  : catB);
  const float* g  = br == 0 ? desG  : (br == 1 ? numG  : catG);
  const float* be = br == 0 ? desBe : (br == 1 ? numBe : catBe);
  const float* m  = br == 0 ? desM  : (br == 1 ? numM  : catM);
  const float* v  = br == 0 ? desV  : (br == 1 ? numV  : catV);
  float a = g[n] * rsqrtf(v[n] + kEps);
  ac[br * kH + n]          = a;
  ac[3 * kH + br * kH + n] = b[n] * a + be[n] - m[n] * a;
}

// ---------------------------------------------------------------------------
// Fused: x = sum of 3 leaky(BN(linear)) branches, then xr[r] = x @ Wr^T.
// One block = 16 node rows. 4 waves; wave w owns output cols [16w,16w+16).
// Phase 1: des GEMM K=768. The 307 MB des stream is staged into LDS with a
//   double-buffered GLOBAL_LOAD_ASYNC_TO_LDS_B128 pipeline (12 chunks of
//   16x64 fp32; issue c+1, s_wait_asynccnt 2, barrier, 16 WMMA K=4 steps).
//   LDS row stride 68 floats -> conflict-free ds_load_b64 fragment reads.
//   desW is tiny (192 KB, L2-resident) and read directly.
// Phase 2: x tile lives in LDS only (never hits HBM); each wave runs two
//   16x16 relation-GEMM tiles (K=64) from LDS.
// Fragment layouts per CDNA5 ISA:
//   A 16x4 : lanes 0-15 rows M, half 0 holds K={0,1}, half 1 K={2,3}
//   B 4x16 : lanes hold col N=(lane&15), halves split K like A
//   C/D    : VGPR j -> M = j + 8*(lane>=16), N = lane&15
// ---------------------------------------------------------------------------
__global__ __launch_bounds__(128) void proj_rgcnw_kernel(
    const float* __restrict__ des, const float* __restrict__ num,
    const float* __restrict__ cat, const float* __restrict__ desW,
    const float* __restrict__ numW, const float* __restrict__ catW,
    const float* __restrict__ rgcnW, const float* __restrict__ ac,
    float* __restrict__ xr) {
  __shared__ float s_a[2][16 * 68];  // double-buffered des K-chunk
  __shared__ float s_x[16 * 66];
  __shared__ float s_num[16 * 5];
  __shared__ float s_cat[16 * 6];

  const int tile = blockIdx.x;
  const int t = threadIdx.x;
  const int rowBase = tile * 16;

  if (t < 80) {
    int i = t / 5, k = t % 5;
    s_num[t] = num[(size_t)(rowBase + i) * 5 + k];
  } else if (t < 176) {
    int u = t - 80, i = u / 6, k = u % 6;
    s_cat[u] = cat[(size_t)(rowBase + i) * 6 + k];
  }

  const int lane = t & 31;
  const int wave = t >> 5;
  const int half = lane >> 4;
  const int l15 = lane & 15;
  const int n = wave * 16 + l15;  // output column 0..63

  const float* brow = desW + (size_t)n * 768;

  // per-lane constants (n is fixed per lane)
  const float Ades = ac[0 * kH + n], Cdes = ac[192 + 0 * kH + n];
  const float Anum = ac[1 * kH + n], Cnum = ac[192 + 1 * kH + n];
  const float Acat = ac[2 * kH + n], Ccat = ac[192 + 2 * kH + n];
  float wn[5], wc[6];
#pragma unroll
  for (int k = 0; k < 5; ++k) wn[k] = numW[n * 5 + k];
#pragma unroll
  for (int k = 0; k < 6; ++k) wc[k] = catW[n * 6 + k];

  // copy role: each thread moves 2 x 16B per chunk (128 thr * 32B = 4 KB)
  const int cprow0 = t >> 4, cpseg0 = t & 15;            // rows 0..7
  const int cprow1 = (t + 128) >> 4, cpseg1 = t & 15;    // rows 8..15
  const float* desRow0 = des + (size_t)(rowBase + cprow0) * 768 + cpseg0 * 4;
  const float* desRow1 = des + (size_t)(rowBase + cprow1) * 768 + cpseg1 * 4;

  // prologue: stage chunk 0 into buffer 0
  copy16_to_lds(desRow0, &s_a[0][cprow0 * 68 + cpseg0 * 4]);
  copy16_to_lds(desRow1, &s_a[0][cprow1 * 68 + cpseg1 * 4]);

  v8f acc = {};
  for (int c = 0; c < kNC; ++c) {
    const int buf = c & 1;
    if (c + 1 < kNC) {  // issue next chunk into the other buffer
      const int kc1 = (c + 1) * kKC;
      copy16_to_lds(desRow0 + kc1, &s_a[buf ^ 1][cprow0 * 68 + cpseg0 * 4]);
      copy16_to_lds(desRow1 + kc1, &s_a[buf ^ 1][cprow1 * 68 + cpseg1 * 4]);
      wait_async_lds<2>();  // chunk c landed (async ops retire in order)
    } else {
      wait_async_lds<0>();
    }
    __syncthreads();  // chunk c visible from all waves; prev buf free

    const int kc = c * kKC;
#pragma unroll
    for (int kb = 0; kb < kKC; kb += 4) {
      int ko = kb + half * 2;
      v2f a = *(const v2f*)&s_a[buf][l15 * 68 + ko];
      v2f b = *(const v2f*)(brow + kc + ko);
      acc = wmma_f32_k4(a, b, acc);
    }
    __syncthreads();  // all waves done with buf before it is refilled
  }

#pragma unroll
  for (int j = 0; j < 8; ++j) {
    int mloc = half * 8 + j;
    float xdes = leaky(acc[j] * Ades + Cdes);
    float dn = 0.0f;
#pragma unroll
    for (int k = 0; k < 5; ++k) dn += s_num[mloc * 5 + k] * wn[k];
    float xnum = leaky(dn * Anum + Cnum);
    float dc = 0.0f;
#pragma unroll
    for (int k = 0; k < 6; ++k) dc += s_cat[mloc * 6 + k] * wc[k];
    float xcat = leaky(dc * Acat + Ccat);
    s_x[mloc * 66 + n] = xdes + xnum + xcat;
  }
  __syncthreads();

  // Phase 2: xr[r] = x @ Wr^T ; 8 tiles (2 relations x 4 col tiles), 2 per wave
#pragma unroll
  for (int tt = 0; tt < 2; ++tt) {
    int flat = wave * 2 + tt;
    int r = flat >> 2;
    int nc = (flat & 3) * 16 + l15;  // output column
    const float* wrow = rgcnW + ((size_t)r * kH + nc) * kH;
    v8f c2 = {};
#pragma unroll
    for (int kb = 0; kb < kH; kb += 4) {
      int ko = kb + half * 2;
      v2f a = *(const v2f*)&s_x[l15 * 66 + ko];
      v2f b = *(const v2f*)(wrow + ko);
      c2 = wmma_f32_k4(a, b, c2);
    }
#pragma unroll
    for (int j = 0; j < 8; ++j) {
      int mloc = half * 8 + j;
      xr[((size_t)r * kN + (rowBase + mloc)) * kH + nc] = c2[j];
    }
  }
}

// ---------------------------------------------------------------------------
// Per-(edge, channel) gather + scatter-add; channel 0 counts the edge.
// 64 consecutive threads share one edge -> fully coalesced 256B gather of
// xr[r][src]; atomics land in L2.
// ---------------------------------------------------------------------------
__global__ __launch_bounds__(256) void edge_scatter_kernel(
    const int* __restrict__ ei, const int* __restrict__ et,
    const float* __restrict__ xr, float* __restrict__ s,
    float* __restrict__ cnt) {
  int t = blockIdx.x * 256 + threadIdx.x;
  int e = t >> 6;
  if (e >= kE) return;
  int c = t & 63;
  int r = et[e];
  int src = ei[e];        // row 0 = source
  int dst = ei[kE + e];   // row 1 = target
  float v = xr[((size_t)r * kN + src) * kH + c];
  atomicAdd(&s[((size_t)r * kN + dst) * kH + c], v);
  if (c == 0) atomicAdd(&cnt[(size_t)r * kN + dst], 1.0f);
}

// ---------------------------------------------------------------------------
// x2 = leaky( (s0/max(c0,1) + s1/max(c1,1)) / R )
// ---------------------------------------------------------------------------
__global__ __launch_bounds__(256) void combine_kernel(
    const float* __restrict__ s, const float* __restrict__ cnt,
    float* __restrict__ x2) {
  int t = blockIdx.x * 256 + threadIdx.x;
  if (t >= kN * kH) return;
  int nnode = t >> 6;
  float c0 = fmaxf(cnt[nnode], 1.0f);
  float c1 = fmaxf(cnt[kN + nnode], 1.0f);
  float acc = s[t] / c0 + s[(size_t)kN * kH + t] / c1;
  x2[t] = leaky(acc * 0.5f);
}

// ---------------------------------------------------------------------------
// Classifier: h = leaky(x2 @ W1^T + b1) via WMMA (K=64, 2 col tiles),
// then out = h @ W2^T + b2 via LDS hand-off (2-wide output on VALU).
// ---------------------------------------------------------------------------
__global__ __launch_bounds__(64) void classifier_kernel(
    const float* __restrict__ x2, const float* __restrict__ W1,
    const float* __restrict__ b1, const float* __restrict__ W2,
    const float* __restrict__ b2, float* __restrict__ out) {
  __shared__ float s_h[16 * 34];
  const int tile = blockIdx.x;
  const int t = threadIdx.x;
  const int lane = t & 31;
  const int wave = t >> 5;  // 0..1
  const int half = lane >> 4;
  const int l15 = lane & 15;
  const int rowBase = tile * 16;
  const int nh = wave * 16 + l15;  // h channel 0..31

  const float* arow = x2 + (size_t)(rowBase + l15) * kH;
  const float* brow = W1 + (size_t)nh * kH;
  const float bias = b1[nh];

  v8f acc = {};
#pragma unroll
  for (int kb = 0; kb < kH; kb += 4) {
    int ko = kb + half * 2;
    v2f a = *(const v2f*)(arow + ko);
    v2f b = *(const v2f*)(brow + ko);
    acc = wmma_f32_k4(a, b, acc);
  }
#pragma unroll
  for (int j = 0; j < 8; ++j) {
    int mloc = half * 8 + j;
    s_h[mloc * 34 + nh] = leaky(acc[j] + bias);
  }
  __syncthreads();
  if (t < 32) {
    int m = t & 15, o = t >> 4;
    float sum = b2[o];
#pragma unroll
    for (int j = 0; j < 32; ++j) sum += s_h[m * 34 + j] * W2[o * 32 + j];
    out[(size_t)(rowBase + m) * 2 + o] = sum;
  }
}

extern "C" void kernel_launch(void* const* d_in, const int* in_sizes, int n_in,
                              void* d_out, int out_size, void* d_ws,
                              size_t ws_size, hipStream_t stream) {
  const float* des  = (const float*)d_in[0];
  const float* num  = (const float*)d_in[1];
  const float* cat  = (const float*)d_in[2];
  const int*   ei   = (const int*)d_in[3];
  const int*   et   = (const int*)d_in[4];
  const float* desW = (const float*)d_in[5];
  const float* desB = (const float*)d_in[6];
  const float* numW = (const float*)d_in[7];
  const float* numB = (const float*)d_in[8];
  const float* catW = (const float*)d_in[9];
  const float* catB = (const float*)d_in[10];
  const float* desG = (const float*)d_in[11];
  const float* desBe= (const float*)d_in[12];
  const float* desM = (const float*)d_in[13];
  const float* desV = (const float*)d_in[14];
  const float* numG = (const float*)d_in[15];
  const float* numBe= (const float*)d_in[16];
  const float* numM = (const float*)d_in[17];
  const float* numV = (const float*)d_in[18];
  const float* catG = (const float*)d_in[19];
  const float* catBe= (const float*)d_in[20];
  const float* catM = (const float*)d_in[21];
  const float* catV = (const float*)d_in[22];
  const float* rgcnW= (const float*)d_in[23];
  const float* W1   = (const float*)d_in[24];
  const float* b1   = (const float*)d_in[25];
  const float* W2   = (const float*)d_in[26];
  const float* b2   = (const float*)d_in[27];

  float* out = (float*)d_out;
  float* ws  = (float*)d_ws;
  float* ac  = ws + WS_AC;
  float* xr  = ws + WS_XR;
  float* s   = ws + WS_S;
  float* cnt = ws + WS_CNT;
  float* x2  = ws + WS_X2;

  bn_fold_kernel<<<1, 256, 0, stream>>>(desB, desG, desBe, desM, desV,
                                        numB, numG, numBe, numM, numV,
                                        catB, catG, catBe, catM, catV, ac);
  // zero segment sums + counts (contiguous region) every call
  (void)hipMemsetAsync(s, 0,
                       (size_t)(2 * kN * kH + 2 * kN) * sizeof(float), stream);
  proj_rgcnw_kernel<<<kNT, 128, 0, stream>>>(des, num, cat, desW, numW, catW,
                                             rgcnW, ac, xr);
  edge_scatter_kernel<<<(kE * 64) / 256, 256, 0, stream>>>(ei, et, xr, s, cnt);
  combine_kernel<<<(kN * kH) / 256, 256, 0, stream>>>(s, cnt, x2);
  classifier_kernel<<<kNT, 64, 0, stream>>>(x2, W1, b1, W2, b2, out);
}